// GraphAttentionLayer_47983374631488
// MI455X (gfx1250) — compile-verified
//
#include <hip/hip_runtime.h>
#include <hip/hip_bf16.h>

typedef __attribute__((ext_vector_type(2))) float v2f;
typedef __attribute__((ext_vector_type(4))) float v4f;
typedef __attribute__((ext_vector_type(8))) float v8f;
typedef __attribute__((ext_vector_type(4))) int   v4i;

#define NV    10000     // N nodes
#define INF_  256       // in_features
#define OUTF  128       // out_features
#define ALPHA 0.2f
#define NTILES 625      // NV / 16

// ---------------------------------------------------------------------------
// Kernel 1: Wh = h @ W   [10000,256]x[256,128], fp32 WMMA 16x16x4.
// One wave per 16-row tile, full 128-col accumulation (8 N-tiles, 64 acc VGPRs)
// ---------------------------------------------------------------------------
__global__ __launch_bounds__(128) void wh_gemm_kernel(
    const float* __restrict__ h, const float* __restrict__ W,
    float* __restrict__ Wh) {
  int wave = threadIdx.x >> 5;
  int lane = threadIdx.x & 31;
  int tile = blockIdx.x * 4 + wave;
  if (tile >= NTILES) return;            // wave-uniform guard (EXEC stays full)
  int m0 = tile * 16;
  int half = lane >> 4;                  // 0: lanes 0-15, 1: lanes 16-31
  int l16  = lane & 15;
  int koff = half * 2;                   // A/B K sub-offset per ISA layout
  const float* hrow = h + (size_t)(m0 + l16) * INF_;

  v8f acc[8] = {};
  for (int k0 = 0; k0 < INF_; k0 += 4) {
    // A 16x4 fp32: v0 = K(koff), v1 = K(koff+1) -> contiguous pair (b64 load)
    v2f av;
    av.x = hrow[k0 + koff];
    av.y = hrow[k0 + koff + 1];
#pragma unroll
    for (int nt = 0; nt < 8; nt++) {
      v2f bv;
      const float* wp = W + (size_t)(k0 + koff) * OUTF + nt * 16 + l16;
      bv.x = wp[0];
      bv.y = wp[OUTF];
      acc[nt] = __builtin_amdgcn_wmma_f32_16x16x4_f32(
          false, av, false, bv, (short)0, acc[nt], false, false);
    }
  }
  // C/D layout: VGPR v -> M = v + half*8, N = l16
#pragma unroll
  for (int nt = 0; nt < 8; nt++) {
#pragma unroll
    for (int v = 0; v < 8; v++) {
      int M = v + half * 8;
      Wh[(size_t)(m0 + M) * OUTF + nt * 16 + l16] = acc[nt][v];
    }
  }
}

// ---------------------------------------------------------------------------
// Kernel 2: s1[i] = Wh[i]·a[0:128] + emb[i]·a[128:256]
//           s2[i] = Wh[i]·a[256:384] + emb[i]·a[384:512]
// One wave32 per row, shuffle reduction.
// ---------------------------------------------------------------------------
__global__ __launch_bounds__(256) void attn_coef_kernel(
    const float* __restrict__ Wh, const float* __restrict__ emb,
    const float* __restrict__ a, float* __restrict__ s1,
    float* __restrict__ s2) {
  int wave = threadIdx.x >> 5;
  int lane = threadIdx.x & 31;
  int row = blockIdx.x * 8 + wave;       // 1250*8 == 10000 exactly
  float p1 = 0.f, p2 = 0.f;
#pragma unroll
  for (int c = lane; c < OUTF; c += 32) {
    float wv = Wh[(size_t)row * OUTF + c];
    float ev = emb[(size_t)row * OUTF + c];
    p1 += wv * a[c]            + ev * a[OUTF + c];
    p2 += wv * a[2 * OUTF + c] + ev * a[3 * OUTF + c];
  }
#pragma unroll
  for (int off = 16; off > 0; off >>= 1) {
    p1 += __shfl_down(p1, off, 32);
    p2 += __shfl_down(p2, off, 32);
  }
  if (lane == 0) { s1[row] = p1; s2[row] = p2; }
}

// ---------------------------------------------------------------------------
// Kernel 3: per 16-row tile:
//   Phase A: online softmax stats (m,l) over masked leaky(s1[i]+s2[j])
//   Phase B: p = exp(e-m)/l, write attention (NT), accumulate h' = p @ Wh WMMA
// 256 threads = 8 waves per block, 625 blocks.
// adj reads and attention writes are NON-TEMPORAL so the 1.2 GB of streaming
// traffic does not evict the L2-resident Wh (5 MB) B-operand working set.
// ---------------------------------------------------------------------------
__global__ __launch_bounds__(256) void attn_softmax_spmm_kernel(
    const int* __restrict__ adj, const float* __restrict__ Wh,
    const float* __restrict__ s1, const float* __restrict__ s2,
    float* __restrict__ hprime, float* __restrict__ att) {
  __shared__ float sm[256], sl[256];
  __shared__ float rowm[16], rowrl[16], rows1[16];
  __shared__ float pbuf[8][256];          // per-wave 16x16 p tile
  __shared__ float hacc[16 * OUTF];       // cross-wave h' reduction

  const int t  = threadIdx.x;
  const int r0 = blockIdx.x * 16;

  // zero h' accumulator
#pragma unroll
  for (int i = 0; i < 8; i++) hacc[i * 256 + t] = 0.f;

  // ---------------- Phase A: row stats (m, l) --------------------------
  {
    int rA = t >> 4;                      // row within tile (16 thr/row)
    int cA = t & 15;
    float s1r = s1[r0 + rA];
    const int* arow = adj + (size_t)(r0 + rA) * NV;
    float m = -3e38f, l = 0.f;
    for (int j0 = 0; j0 < NV; j0 += 16) {
      __builtin_prefetch(arow + j0 + 1024, 0, 0);   // global_prefetch_b8
      int av = __builtin_nontemporal_load(arow + j0 + cA);  // NT stream
      float e = s1r + s2[j0 + cA];
      e = e > 0.f ? e : ALPHA * e;
      if (av > 0) {
        float mn = fmaxf(m, e);
        l = l * __expf(m - mn) + __expf(e - mn);
        m = mn;
      }
    }
    sm[t] = m; sl[t] = l;
  }
  __syncthreads();
  if (t < 16) {
    float M = -3e38f, L = 0.f;
#pragma unroll
    for (int i = 0; i < 16; i++) {
      float mi = sm[t * 16 + i], li = sl[t * 16 + i];
      float mn = fmaxf(M, mi);
      L = L * __expf(M - mn) + li * __expf(mi - mn);
      M = mn;
    }
    rowm[t]  = M;
    rowrl[t] = (L > 0.f) ? 1.f / L : 0.f;
    rows1[t] = s1[r0 + t];
  }
  __syncthreads();

  // ---------------- Phase B: attention + WMMA accumulate ---------------
  const int wave = t >> 5;
  const int lane = t & 31;
  const int half = lane >> 4;
  const int l16  = lane & 15;
  const int koff = half * 2;
  float* pw = pbuf[wave];

  v8f acc[8] = {};
  for (int c = wave; c < NTILES; c += 8) {     // wave-uniform loop
    int j0 = c * 16;
    int r  = lane >> 1;                        // row 0..15
    int cb = (lane & 1) * 8;                   // col base 0 or 8
    const int* ap = adj + (size_t)(r0 + r) * NV + j0 + cb;
    v4i q0 = __builtin_nontemporal_load((const v4i*)ap);        // NT b128
    v4i q1 = __builtin_nontemporal_load((const v4i*)(ap + 4));  // NT b128
    int aval[8] = {q0.x, q0.y, q0.z, q0.w, q1.x, q1.y, q1.z, q1.w};

    float mrow = rowm[r], rl = rowrl[r], s1v = rows1[r];
    float p[8];
#pragma unroll
    for (int i = 0; i < 8; i++) {
      float e = s1v + s2[j0 + cb + i];
      e = e > 0.f ? e : ALPHA * e;
      p[i] = (aval[i] > 0) ? __expf(e - mrow) * rl : 0.f;
    }
    v4f pv0 = {p[0], p[1], p[2], p[3]};
    v4f pv1 = {p[4], p[5], p[6], p[7]};
    // write final attention: non-temporal b128 stores (write-once stream)
    float* op = att + (size_t)(r0 + r) * NV + j0 + cb;
    __builtin_nontemporal_store(pv0, (v4f*)op);
    __builtin_nontemporal_store(pv1, (v4f*)(op + 4));
    // stage p tile to LDS for WMMA A-operand gather
    *(v4f*)(pw + r * 16 + cb)     = pv0;
    *(v4f*)(pw + r * 16 + cb + 4) = pv1;

    // D += A(16x4 of p) * B(4x16 of Wh), K-chunks of 4 within the 16-j tile
#pragma unroll
    for (int kk = 0; kk < 16; kk += 4) {
      v2f av;
      av.x = pw[l16 * 16 + kk + koff];         // ds_load (b64 pair)
      av.y = pw[l16 * 16 + kk + koff + 1];
#pragma unroll
      for (int nt = 0; nt < 8; nt++) {
        v2f bv;
        const float* wp = Wh + (size_t)(j0 + kk + koff) * OUTF + nt * 16 + l16;
        bv.x = wp[0];                          // L2-resident Wh
        bv.y = wp[OUTF];
        acc[nt] = __builtin_amdgcn_wmma_f32_16x16x4_f32(
            false, av, false, bv, (short)0, acc[nt], false, false);
      }
    }
  }

  // cross-wave reduction of partial h' tiles via LDS float atomics
#pragma unroll
  for (int nt = 0; nt < 8; nt++) {
#pragma unroll
    for (int v = 0; v < 8; v++) {
      int M = v + half * 8;
      atomicAdd(&hacc[M * OUTF + nt * 16 + l16], acc[nt][v]);   // ds_add_f32
    }
  }
  __syncthreads();

  // tile rows are contiguous in h' -> flat coalesced store of 2048 floats
  v4f* hp4 = (v4f*)(hprime + (size_t)r0 * OUTF);
  const v4f* ha4 = (const v4f*)hacc;
#pragma unroll
  for (int i = 0; i < 2; i++) hp4[i * 256 + t] = ha4[i * 256 + t];
}

// ---------------------------------------------------------------------------
extern "C" void kernel_launch(void* const* d_in, const int* in_sizes, int n_in,
                              void* d_out, int out_size, void* d_ws, size_t ws_size,
                              hipStream_t stream) {
  const float* h   = (const float*)d_in[0];
  const int*   adj = (const int*)d_in[1];
  const float* emb = (const float*)d_in[2];
  const float* W   = (const float*)d_in[3];
  const float* a   = (const float*)d_in[4];

  float* hprime = (float*)d_out;                      // [10000,128]
  float* att    = (float*)d_out + (size_t)NV * OUTF;  // [10000,10000]

  float* Wh = (float*)d_ws;                           // 10000*128 floats
  float* s1 = Wh + (size_t)NV * OUTF;                 // 10000
  float* s2 = s1 + NV;                                // 10000

  wh_gemm_kernel<<<(NTILES + 3) / 4, 128, 0, stream>>>(h, W, Wh);
  attn_coef_kernel<<<NV / 8, 256, 0, stream>>>(Wh, emb, a, s1, s2);
  attn_softmax_spmm_kernel<<<NTILES, 256, 0, stream>>>(adj, Wh, s1, s2, hprime, att);
}